// SubsamplingRoiLayer_80135499809398
// MI455X (gfx1250) — compile-verified
//
#include <hip/hip_runtime.h>
#include <math.h>
#include <stdint.h>

// Problem constants (from the reference)
#define R_TOTAL      20000
#define G_TOTAL      256
#define N_SAMPLES    64
#define POS_PER_BATCH 16
#define EPS          1e-7f
#define INV_IMG      (1.0f / 1024.0f)
#define BPI          ((R_TOTAL + 255) / 256)   // blocks per image = 79

// ---------------------------------------------------------------------------
// Kernel 1: per-RoI max-IoU / argmax over 256 GT boxes + pos/neg classification.
// GT tile (4KB) is staged into LDS via the CDNA5 async-to-LDS data path so the
// hot loop reads uniform-address ds_load_b128 broadcasts. Division-free argmax.
// Output: packed int per (b, r): (flag << 16) | argmax_gt, flag: 0/1(pos)/2(neg)
// ---------------------------------------------------------------------------
__global__ __launch_bounds__(256) void iou_argmax_kernel(
    const float* __restrict__ rois,   // (B, R, 4) normalized
    const float* __restrict__ gts,    // (B, G, 4) raw pixel coords
    int* __restrict__ packed_out)     // (B, R)
{
  __shared__ float4 s_box[G_TOTAL];   // normalized GT boxes
  __shared__ float  s_area[G_TOTAL];  // GT area; -1.0f sentinel = invalid GT

  const int tid = threadIdx.x;
  const int b   = blockIdx.x / BPI;
  const int r   = (blockIdx.x % BPI) * 256 + tid;

  // --- CDNA5 async copy: global -> LDS, one b128 per thread (256*16B = 4KB) ---
  const float* gt_img = gts + (size_t)b * G_TOTAL * 4;
  {
    unsigned lds_dst = (unsigned)(unsigned long long)(uintptr_t)&s_box[tid];
    unsigned goff    = (unsigned)tid * 16u;
    asm volatile("global_load_async_to_lds_b128 %0, %1, %2"
                 :: "v"(lds_dst), "v"(goff), "s"(gt_img)
                 : "memory");
  }

  // Load this thread's RoI while the async copy is in flight.
  float4 roi = make_float4(0.f, 0.f, 0.f, 0.f);
  const bool r_ok = (r < R_TOTAL);
  if (r_ok) roi = *(const float4*)(rois + ((size_t)b * R_TOTAL + r) * 4);

  asm volatile("s_wait_asynccnt 0" ::: "memory");
  __syncthreads();

  // Normalize staged boxes in place; precompute area (+validity sentinel).
  {
    float4 raw = s_box[tid];
    bool gv = (raw.x != 0.f) | (raw.y != 0.f) | (raw.z != 0.f) | (raw.w != 0.f);
    float4 nb = make_float4(raw.x * INV_IMG, raw.y * INV_IMG,
                            raw.z * INV_IMG, raw.w * INV_IMG);
    s_box[tid]  = nb;
    float ag    = fmaxf(nb.z - nb.x, 0.f) * fmaxf(nb.w - nb.y, 0.f);
    s_area[tid] = gv ? ag : -1.0f;
  }
  __syncthreads();

  const bool roi_valid =
      (roi.x != 0.f) | (roi.y != 0.f) | (roi.z != 0.f) | (roi.w != 0.f);
  const float area_r = fmaxf(roi.z - roi.x, 0.f) * fmaxf(roi.w - roi.y, 0.f);

  // Division-free running argmax over iou = inter/union (union > 0 always).
  // Invalid GT contributes exactly iou = -1 (represented as i=-1, u=1).
  float best_i = -1.0f, best_u = 1.0f;
  int   best_g = 0;
#pragma unroll 8
  for (int g = 0; g < G_TOTAL; ++g) {
    float4 gb = s_box[g];         // uniform-address LDS broadcast
    float  ag = s_area[g];
    float yy1 = fmaxf(roi.x, gb.x);
    float xx1 = fmaxf(roi.y, gb.y);
    float yy2 = fminf(roi.z, gb.z);
    float xx2 = fminf(roi.w, gb.w);
    float inter = fmaxf(yy2 - yy1, 0.f) * fmaxf(xx2 - xx1, 0.f);
    float un    = fmaxf(area_r + ag - inter, EPS);
    bool  inv   = (ag < 0.f);
    float ci = inv ? -1.0f : inter;
    float cu = inv ?  1.0f : un;
    // iou_g > best_iou  <=>  ci*best_u > best_i*cu   (strict: first max wins)
    if (ci * best_u > best_i * cu) { best_i = ci; best_u = cu; best_g = g; }
  }

  // Classify (thresholds done in cross-multiplied form; best_u > 0).
  bool ge_pos = (best_i >= 0.5f * best_u);
  bool ge_exc = (best_i >= 0.1f * best_u);
  bool pos = roi_valid && ge_pos;
  bool neg = roi_valid && !ge_pos && ge_exc;
  int flag = pos ? 1 : (neg ? 2 : 0);

  if (r_ok) packed_out[(size_t)b * R_TOTAL + r] = (flag << 16) | best_g;
}

// ---------------------------------------------------------------------------
// Kernel 2: one wave32 per image. Ballot-based stream compaction gathers the
// lowest-index 16 positives and 64 negatives in index order (matches the
// reference's deterministic top_k on (R - idx) keys), then emits samples.
// ---------------------------------------------------------------------------
__global__ __launch_bounds__(32) void select_emit_kernel(
    const float* __restrict__ rois,
    const float* __restrict__ gts,
    const int*   __restrict__ labels,   // (B, G) int32
    const int*   __restrict__ packed,   // (B, R)
    float* __restrict__ out,            // [roi | off | lab] concatenated
    int B)
{
  const int b    = blockIdx.x;
  const int lane = threadIdx.x;

  __shared__ int s_pos[POS_PER_BATCH];
  __shared__ int s_neg[N_SAMPLES];

  const int* pk_img = packed + (size_t)b * R_TOTAL;
  const unsigned lt_mask = (1u << lane) - 1u;

  int pc = 0, nc = 0;   // uniform (derived from ballots) -> live in SGPRs
#pragma unroll 5
  for (int base = 0; base < R_TOTAL; base += 32) {   // 625 iterations, no break
    int pkv  = pk_img[base + lane];
    int flag = pkv >> 16;
    unsigned pm = __builtin_amdgcn_ballot_w32(flag == 1);
    unsigned nm = __builtin_amdgcn_ballot_w32(flag == 2);
    if (flag == 1) {
      int s = pc + __popc(pm & lt_mask);
      if (s < POS_PER_BATCH) s_pos[s] = base + lane;
    }
    if (flag == 2) {
      int s = nc + __popc(nm & lt_mask);
      if (s < N_SAMPLES) s_neg[s] = base + lane;
    }
    pc += __popc(pm);
    nc += __popc(nm);
  }
  __syncthreads();

  const int n_pos = (pc < POS_PER_BATCH) ? pc : POS_PER_BATCH;
  const int cap   = N_SAMPLES - n_pos;
  const int n_neg = (nc < cap) ? nc : cap;

  const size_t roi_base = 0;
  const size_t off_base = (size_t)B * N_SAMPLES * 4;
  const size_t lab_base = 2 * (size_t)B * N_SAMPLES * 4;

  for (int s = lane; s < N_SAMPLES; s += 32) {
    bool isp = (s < n_pos);
    bool isn = !isp && (s < n_pos + n_neg);
    float o0 = 0.f, o1 = 0.f, o2 = 0.f, o3 = 0.f;
    float f0 = 0.f, f1 = 0.f, f2 = 0.f, f3 = 0.f;
    float lab = 0.f;
    if (isp || isn) {
      int ridx = isp ? s_pos[s] : s_neg[s - n_pos];
      int gi   = pk_img[ridx] & 0xFFFF;
      const float* rp = rois + ((size_t)b * R_TOTAL + ridx) * 4;
      float r0 = rp[0], r1 = rp[1], r2 = rp[2], r3 = rp[3];
      const float* gp = gts + ((size_t)b * G_TOTAL + gi) * 4;
      float g0 = gp[0] * INV_IMG, g1 = gp[1] * INV_IMG;
      float g2 = gp[2] * INV_IMG, g3 = gp[3] * INV_IMG;
      o0 = r0; o1 = r1; o2 = r2; o3 = r3;
      float rh  = fmaxf(r2 - r0, EPS), rw  = fmaxf(r3 - r1, EPS);
      float rcy = r0 + 0.5f * rh,      rcx = r1 + 0.5f * rw;
      float gh  = fmaxf(g2 - g0, EPS), gw  = fmaxf(g3 - g1, EPS);
      float gcy = g0 + 0.5f * gh,      gcx = g1 + 0.5f * gw;
      f0 = ((gcy - rcy) / rh) / 0.1f;
      f1 = ((gcx - rcx) / rw) / 0.1f;
      f2 = logf(gh / rh) / 0.2f;
      f3 = logf(gw / rw) / 0.2f;
      if (isp) lab = (float)labels[b * G_TOTAL + gi];
    }
    float* ro = out + roi_base + ((size_t)b * N_SAMPLES + s) * 4;
    ro[0] = o0; ro[1] = o1; ro[2] = o2; ro[3] = o3;
    float* fo = out + off_base + ((size_t)b * N_SAMPLES + s) * 4;
    fo[0] = f0; fo[1] = f1; fo[2] = f2; fo[3] = f3;
    out[lab_base + (size_t)b * N_SAMPLES + s] = lab;
  }
}

extern "C" void kernel_launch(void* const* d_in, const int* in_sizes, int n_in,
                              void* d_out, int out_size, void* d_ws, size_t ws_size,
                              hipStream_t stream) {
  const float* rois   = (const float*)d_in[0];  // (B, 20000, 4) f32
  const float* gts    = (const float*)d_in[1];  // (B, 256, 4)  f32
  const int*   labels = (const int*)d_in[2];    // (B, 256)     i32
  float*       out    = (float*)d_out;

  const int B = in_sizes[0] / (R_TOTAL * 4);
  int* packed = (int*)d_ws;                     // B * 20000 ints

  iou_argmax_kernel<<<B * BPI, 256, 0, stream>>>(rois, gts, packed);
  select_emit_kernel<<<B, 32, 0, stream>>>(rois, gts, labels, packed, out, B);
}